// MyInstanceNorm_80788334838451
// MI455X (gfx1250) — compile-verified
//
#include <hip/hip_runtime.h>
#include <stdint.h>

typedef float v4f __attribute__((ext_vector_type(4)));  // real clang vector

// Problem constants (from reference: x shape (8,16,2000,128), POOL=10)
constexpr int B_ = 8, C_ = 16, T_ = 2000, F_ = 128;
constexpr int SL = B_ * C_;          // 128 independent (b,c) slices
constexpr int POOL = 10, T1 = T_ / POOL;  // 200 pooled blocks
constexpr float EPS = 1e-5f;

// K1 tiling: each WG covers 20 pool-blocks (200 timesteps) of one slice,
// streamed as 5 chunks of 4 pool-blocks (40 t x 128 f = 20480 B) through a
// double-buffered LDS tile filled with async global->LDS B128 copies.
constexpr int BLKS_PER_WG = 20;
constexpr int GROUPS = T1 / BLKS_PER_WG;   // 10 groups per slice
constexpr int CHUNK_BLKS = 4;
constexpr int CHUNK_T = CHUNK_BLKS * POOL; // 40
constexpr int NCHUNK = BLKS_PER_WG / CHUNK_BLKS; // 5
constexpr int CHUNK_BYTES = CHUNK_T * F_ * 4;    // 20480
constexpr int LOADS_PER_THREAD = CHUNK_BYTES / (F_ * 16); // 10 x b128

__device__ __forceinline__ uint32_t lds_addr_of(const void* p) {
    // Low 32 bits of the flat shared-aperture address are the LDS byte offset
    // (ISA 10.2).
    return (uint32_t)(uintptr_t)p;
}

__device__ __forceinline__ void async_copy_chunk(const float* __restrict__ gsrc,
                                                 uint32_t lds_base, int tid) {
    const char* g = (const char*)gsrc + (size_t)tid * 16;
    uint32_t l = lds_base + (uint32_t)tid * 16;
#pragma unroll
    for (int i = 0; i < LOADS_PER_THREAD; ++i) {
        asm volatile("global_load_async_to_lds_b128 %0, %1, off"
                     :: "v"(l), "v"(g) : "memory");
        g += F_ * 16;   // 2048 B: next row group, stays contiguous
        l += F_ * 16;
    }
}

// ---------------- K1: pooled block sums (pm, pm2) ----------------
__global__ void __launch_bounds__(128)
minorm_k1_stats(const float* __restrict__ x,
                float* __restrict__ pm, float* __restrict__ pm2) {
    __shared__ __align__(16) float tile[2][CHUNK_T * F_];  // 2 x 20 KB
    const int s = blockIdx.x / GROUPS;
    const int g = blockIdx.x % GROUPS;
    const int tid = threadIdx.x;          // == f channel

    const float* src = x + (size_t)s * T_ * F_
                         + (size_t)g * BLKS_PER_WG * POOL * F_;
    const uint32_t lds0 = lds_addr_of(&tile[0][0]);
    const uint32_t lds1 = lds_addr_of(&tile[1][0]);

    async_copy_chunk(src, lds0, tid);

    for (int ch = 0; ch < NCHUNK; ++ch) {
        if (ch + 1 < NCHUNK) {
            async_copy_chunk(src + (size_t)(ch + 1) * CHUNK_T * F_,
                             ((ch + 1) & 1) ? lds1 : lds0, tid);
            // 10 new loads in flight; waiting to <=10 retires the previous
            // chunk's 10 loads (async loads complete in order per wave).
            asm volatile("s_wait_asynccnt 10" ::: "memory");
        } else {
            asm volatile("s_wait_asynccnt 0" ::: "memory");
        }
        __syncthreads();  // all waves' fills of current buffer visible

        const float* tl = &tile[ch & 1][0];
#pragma unroll
        for (int b = 0; b < CHUNK_BLKS; ++b) {
            float sum = 0.f, sum2 = 0.f;
#pragma unroll
            for (int j = 0; j < POOL; ++j) {
                float v = tl[(b * POOL + j) * F_ + tid];  // conflict-free
                sum += v; sum2 += v * v;
            }
            const int t1 = g * BLKS_PER_WG + ch * CHUNK_BLKS + b;
            const size_t o = ((size_t)s * T1 + t1) * F_ + tid;
            pm[o]  = sum  * 0.1f;
            pm2[o] = sum2 * 0.1f;
        }
        __syncthreads();  // done reading buffer before it is refilled
    }
}

// ---------------- K2: causal scan -> mean, inv_std (in place) ----------------
__global__ void __launch_bounds__(128)
minorm_k2_scan(float* __restrict__ pm, float* __restrict__ pm2) {
    const int s = blockIdx.x;
    const int f = threadIdx.x;
    float cs = 0.f, cs2 = 0.f;
    size_t o = (size_t)s * T1 * F_ + f;
    for (int i = 0; i < T1; ++i, o += F_) {
        cs  += pm[o];
        cs2 += pm2[o];
        const float inv_n = 1.0f / (float)(i + 1);
        const float mean  = cs * inv_n;
        const float var   = cs2 * inv_n - mean * mean;
        pm[o]  = mean;
        pm2[o] = rsqrtf(var + EPS);
    }
}

// ---------------- K3: fully parallel normalize (v4f, per-block stats) --------
// One thread owns one (slice, pool-block t1, f4-quad): loads mean/istd once,
// then streams 10 timesteps. Wave = 32 lanes over f4 -> every x/out access is
// a contiguous 512 B transaction. blockDim=256 covers 8 t1 rows.
// x loads are non-temporal (last use); out stores are non-temporal (write-once)
// so the 131 MB output stream does not evict L2-resident stats / x lines.
constexpr int F4 = F_ / 4;                 // 32 vector lanes per row
constexpr int T1_PER_BLOCK = 256 / F4;     // 8 pool-blocks per workgroup
__global__ void __launch_bounds__(256)
minorm_k3_norm(const float* __restrict__ x,
               const float* __restrict__ mean_s,
               const float* __restrict__ istd_s,
               float* __restrict__ out) {
    const int s   = blockIdx.y;
    const int t1  = blockIdx.x * T1_PER_BLOCK + (threadIdx.x >> 5); // /32
    const int f4  = threadIdx.x & (F4 - 1);

    const size_t so = ((size_t)s * T1 + t1) * F4 + f4;
    const v4f m  = reinterpret_cast<const v4f*>(mean_s)[so];
    const v4f is = reinterpret_cast<const v4f*>(istd_s)[so];

    // v4f index of first element of this pool block
    size_t idx = ((size_t)s * T_ + (size_t)t1 * POOL) * F4 + f4;
#pragma unroll
    for (int j = 0; j < POOL; ++j, idx += F4) {
        const v4f xv =
            __builtin_nontemporal_load(reinterpret_cast<const v4f*>(x) + idx);
        const v4f o = (xv - m) * is;
        __builtin_nontemporal_store(o, reinterpret_cast<v4f*>(out) + idx);
    }
}

// ---------------- Fallback: fused single-pass (if ws too small) --------------
__global__ void __launch_bounds__(128)
minorm_fused(const float* __restrict__ x, float* __restrict__ out) {
    const int s = blockIdx.x;
    const int f = threadIdx.x;
    const float* xs = x + (size_t)s * T_ * F_ + f;
    float* os = out + (size_t)s * T_ * F_ + f;
    float cs = 0.f, cs2 = 0.f;
    for (int i = 0; i < T1; ++i) {
        float v[POOL];
        float sum = 0.f, sum2 = 0.f;
#pragma unroll
        for (int j = 0; j < POOL; ++j) {
            v[j] = xs[(size_t)(i * POOL + j) * F_];  // coalesced across lanes
            sum += v[j]; sum2 += v[j] * v[j];
        }
        cs  += sum  * 0.1f;
        cs2 += sum2 * 0.1f;
        const float inv_n = 1.0f / (float)(i + 1);
        const float mean  = cs * inv_n;
        const float var   = cs2 * inv_n - mean * mean;
        const float istd  = rsqrtf(var + EPS);
#pragma unroll
        for (int j = 0; j < POOL; ++j)
            os[(size_t)(i * POOL + j) * F_] = (v[j] - mean) * istd;
    }
}

extern "C" void kernel_launch(void* const* d_in, const int* in_sizes, int n_in,
                              void* d_out, int out_size, void* d_ws, size_t ws_size,
                              hipStream_t stream) {
    const float* x = (const float*)d_in[0];
    float* out = (float*)d_out;

    const size_t statsN = (size_t)SL * T1 * F_;          // 3,276,800 floats
    const size_t need   = 2 * statsN * sizeof(float);    // ~26.2 MB

    if (ws_size >= need) {
        float* pm  = (float*)d_ws;
        float* pm2 = pm + statsN;
        minorm_k1_stats<<<SL * GROUPS, 128, 0, stream>>>(x, pm, pm2);
        minorm_k2_scan <<<SL, 128, 0, stream>>>(pm, pm2);
        dim3 grid3(T1 / T1_PER_BLOCK, SL);               // (25, 128)
        minorm_k3_norm <<<grid3, 256, 0, stream>>>(x, pm, pm2, out);
    } else {
        minorm_fused<<<SL, 128, 0, stream>>>(x, out);
    }
}